// GTLayer_54460185313549
// MI455X (gfx1250) — compile-verified
//
#include <hip/hip_runtime.h>
#include <math.h>

#define DFEAT   128
#define NHEADS  8
#define NREL    9
#define NGRAPHS 64

typedef __attribute__((ext_vector_type(2))) float v2f;
typedef __attribute__((ext_vector_type(8))) float v8f;

// ---------------------------------------------------------------------------
// WMMA GEMM:  C[M x NC] = act( A[M x K] @ B[K x NC]  (+ bias | + C_prev) )
// block = 256 threads (8 wave32s). Block covers 16 rows x 128 cols
// (blockIdx.y selects 128-col group for NC=256). Wave w owns 16x16 tile.
// Uses V_WMMA_F32_16X16X4_F32 (fp32, matches reference numerics).
// ---------------------------------------------------------------------------
template <bool ACCUM, bool BIAS, bool RELU>
__global__ __launch_bounds__(256) void gemm_wmma(
    const float* __restrict__ A, const float* __restrict__ B,
    const float* __restrict__ bias, float* __restrict__ C,
    int M, int K, int NC) {
  __shared__ float As[16 * 260];           // K <= 256, stride K+4 (bank-safe)
  const int lds     = K + 4;
  const int rowBase = blockIdx.x * 16;
  const int wave    = threadIdx.x >> 5;
  const int lane    = threadIdx.x & 31;
  const int colBase = blockIdx.y * 128 + wave * 16;

  // cooperative load of the 16 x K A-tile
  for (int i = threadIdx.x; i < 16 * K; i += 256) {
    int r = i / K, cc = i - r * K;
    int row = rowBase + r;
    As[r * lds + cc] = (row < M) ? A[(size_t)row * K + cc] : 0.f;
  }
  __syncthreads();

  const int l15   = lane & 15;
  const int khalf = (lane >> 4) * 2;       // lanes 0-15 -> K0/K1, 16-31 -> K2/K3
  const int col   = colBase + l15;
  const int rhalf = (lane >> 4) << 3;      // C/D: upper half-lanes hold M+8

  v8f c = {};
  if (ACCUM) {
#pragma unroll
    for (int i = 0; i < 8; ++i) {
      int row = rowBase + i + rhalf;
      c[i] = (row < M) ? C[(size_t)row * NC + col] : 0.f;
    }
  } else if (BIAS) {
    float bv = bias[col];
#pragma unroll
    for (int i = 0; i < 8; ++i) c[i] = bv;
  }

  for (int k = 0; k < K; k += 4) {
    v2f a, b;
    a.x = As[l15 * lds + k + khalf];
    a.y = As[l15 * lds + k + khalf + 1];
    b.x = B[(size_t)(k + khalf) * NC + col];
    b.y = B[(size_t)(k + khalf + 1) * NC + col];
    c = __builtin_amdgcn_wmma_f32_16x16x4_f32(false, a, false, b, (short)0, c,
                                              false, false);
  }

#pragma unroll
  for (int i = 0; i < 8; ++i) {
    int row = rowBase + i + rhalf;
    if (row < M) {
      float v = c[i];
      if (RELU) v = fmaxf(v, 0.f);
      C[(size_t)row * NC + col] = v;
    }
  }
}

// ---------------------------------------------------------------------------
// W[r] = sum_b coeff[r,b] * basis[b]   (9 x 128 x 128, tiny)
// ---------------------------------------------------------------------------
__global__ void build_W(const float* __restrict__ coeff,
                        const float* __restrict__ basis,
                        float* __restrict__ W) {
  int t = blockIdx.x * blockDim.x + threadIdx.x;
  if (t >= NREL * DFEAT * DFEAT) return;
  int r  = t / (DFEAT * DFEAT);
  int ij = t - r * DFEAT * DFEAT;
  float acc = 0.f;
#pragma unroll
  for (int b = 0; b < NREL; ++b)
    acc = fmaf(coeff[r * NREL + b], basis[b * DFEAT * DFEAT + ij], acc);
  W[t] = acc;
}

// ---------------------------------------------------------------------------
// GraphNorm (3 passes: stats, sub+var, finalize)
// ---------------------------------------------------------------------------
__global__ void gn_stats(const float* __restrict__ x, const int* __restrict__ seg,
                         float* __restrict__ sums, float* __restrict__ counts,
                         int n) {
  int t = blockIdx.x * blockDim.x + threadIdx.x;
  if (t >= n * DFEAT) return;
  int node = t >> 7, d = t & 127;
  int g = seg[node];
  atomicAdd(&sums[g * DFEAT + d], x[t]);
  if (d == 0) atomicAdd(&counts[g], 1.f);
}

__global__ void gn_sub(const float* __restrict__ x, const int* __restrict__ seg,
                       const float* __restrict__ sums,
                       const float* __restrict__ counts,
                       const float* __restrict__ ms, float* __restrict__ sub,
                       float* __restrict__ varsum, int n) {
  int t = blockIdx.x * blockDim.x + threadIdx.x;
  if (t >= n * DFEAT) return;
  int node = t >> 7, d = t & 127;
  int g = seg[node];
  float mean = sums[g * DFEAT + d] / counts[g];
  float s = x[t] - mean * ms[d];
  sub[t] = s;
  atomicAdd(&varsum[g * DFEAT + d], s * s);
}

__global__ void gn_fin(const float* __restrict__ sub, const int* __restrict__ seg,
                       const float* __restrict__ varsum,
                       const float* __restrict__ counts,
                       const float* __restrict__ w, const float* __restrict__ b,
                       float* __restrict__ out, int n) {
  int t = blockIdx.x * blockDim.x + threadIdx.x;
  if (t >= n * DFEAT) return;
  int node = t >> 7, d = t & 127;
  int g = seg[node];
  float inv = rsqrtf(varsum[g * DFEAT + d] / counts[g] + 1e-6f);
  out[t] = w[d] * sub[t] * inv + b[d];
}

// ---------------------------------------------------------------------------
// S_r[dst] += hn[src] for edges of relation r  (thread = edge x float4-chunk)
// ---------------------------------------------------------------------------
__global__ void scatter_rel(const float* __restrict__ hn,
                            const int* __restrict__ src,
                            const int* __restrict__ dst,
                            const int* __restrict__ et, int r,
                            float* __restrict__ S, int E) {
  int t = blockIdx.x * blockDim.x + threadIdx.x;
  int e = t >> 5, c = t & 31;
  if (e >= E || et[e] != r) return;
  float4 v = ((const float4*)(hn + (size_t)src[e] * DFEAT))[c];
  float* o = S + (size_t)dst[e] * DFEAT + (size_t)c * 4;
  atomicAdd(o + 0, v.x);
  atomicAdd(o + 1, v.y);
  atomicAdd(o + 2, v.z);
  atomicAdd(o + 3, v.w);
}

// ---------------------------------------------------------------------------
// Edge attention: per (edge, head): score = exp(clip(<K[src],Q[dst]>/4))
// wV[dst] += V[src]*score ; z[dst,h] += score
// ---------------------------------------------------------------------------
__global__ void edge_attn(const float* __restrict__ Q, const float* __restrict__ Kf,
                          const float* __restrict__ V, const int* __restrict__ src,
                          const int* __restrict__ dst, float* __restrict__ wV,
                          float* __restrict__ z, int E) {
  int t = blockIdx.x * blockDim.x + threadIdx.x;
  int e = t >> 3, h = t & 7;
  if (e >= E) return;
  int s = src[e], d = dst[e];
  const float4* kp = (const float4*)(Kf + (size_t)s * DFEAT + h * 16);
  const float4* qp = (const float4*)(Q + (size_t)d * DFEAT + h * 16);
  float acc = 0.f;
#pragma unroll
  for (int i = 0; i < 4; ++i) {
    float4 kv = kp[i], qv = qp[i];
    acc += kv.x * qv.x + kv.y * qv.y + kv.z * qv.z + kv.w * qv.w;
  }
  acc = fminf(fmaxf(acc * 0.25f, -10.f), 10.f);  // /sqrt(16), clip
  float sc = expf(acc);
  const float* vp = V + (size_t)s * DFEAT + h * 16;
  float* op = wV + (size_t)d * DFEAT + h * 16;
#pragma unroll
  for (int j = 0; j < 16; ++j) atomicAdd(&op[j], vp[j] * sc);
  atomicAdd(&z[(size_t)d * NHEADS + h], sc);
}

__global__ void attn_norm(float* __restrict__ wV, const float* __restrict__ z,
                          int n) {
  int t = blockIdx.x * blockDim.x + threadIdx.x;
  if (t >= n * DFEAT) return;
  int node = t >> 7, d = t & 127, h = d >> 4;
  wV[t] = wV[t] / (z[(size_t)node * NHEADS + h] + 1e-6f);
}

// ---------------------------------------------------------------------------
// LayerNorm: one wave32 per node, 4 features per lane, shuffle reduction
// ---------------------------------------------------------------------------
__global__ __launch_bounds__(256) void layer_norm_k(
    const float* __restrict__ x, const float* __restrict__ g,
    const float* __restrict__ b, float* __restrict__ out, int n) {
  int node = blockIdx.x * 8 + (threadIdx.x >> 5);
  int lane = threadIdx.x & 31;
  if (node >= n) return;
  float4 v = ((const float4*)(x + (size_t)node * DFEAT))[lane];
  float s = v.x + v.y + v.z + v.w;
#pragma unroll
  for (int o = 16; o > 0; o >>= 1) s += __shfl_xor(s, o, 32);
  float mu = s * (1.f / DFEAT);
  float cx = v.x - mu, cy = v.y - mu, cz = v.z - mu, cw = v.w - mu;
  float s2 = cx * cx + cy * cy + cz * cz + cw * cw;
#pragma unroll
  for (int o = 16; o > 0; o >>= 1) s2 += __shfl_xor(s2, o, 32);
  float inv = rsqrtf(s2 * (1.f / DFEAT) + 1e-5f);
  float4 gv = ((const float4*)g)[lane];
  float4 bv = ((const float4*)b)[lane];
  float4 ov;
  ov.x = gv.x * cx * inv + bv.x;
  ov.y = gv.y * cy * inv + bv.y;
  ov.z = gv.z * cz * inv + bv.z;
  ov.w = gv.w * cw * inv + bv.w;
  ((float4*)(out + (size_t)node * DFEAT))[lane] = ov;
}

// ---------------------------------------------------------------------------
extern "C" void kernel_launch(void* const* d_in, const int* in_sizes, int n_in,
                              void* d_out, int out_size, void* d_ws,
                              size_t ws_size, hipStream_t stream) {
  const float* h      = (const float*)d_in[0];
  const int* src      = (const int*)d_in[1];
  const int* dst      = (const int*)d_in[2];
  const int* etypes   = (const int*)d_in[3];
  const int* seg      = (const int*)d_in[4];
  // d_in[5] = num_graphs (device scalar) -> compile-time NGRAPHS
  const float* q_coeff = (const float*)d_in[6];
  const float* q_basis = (const float*)d_in[7];
  const float* q_loop  = (const float*)d_in[8];
  const float* q_bias  = (const float*)d_in[9];
  const float* k_coeff = (const float*)d_in[10];
  const float* k_basis = (const float*)d_in[11];
  const float* k_loop  = (const float*)d_in[12];
  const float* k_bias  = (const float*)d_in[13];
  const float* v_coeff = (const float*)d_in[14];
  const float* v_basis = (const float*)d_in[15];
  const float* v_loop  = (const float*)d_in[16];
  const float* v_bias  = (const float*)d_in[17];
  const float* o_w     = (const float*)d_in[18];
  const float* o_b     = (const float*)d_in[19];
  const float* gn1_w   = (const float*)d_in[20];
  const float* gn1_b   = (const float*)d_in[21];
  const float* gn1_ms  = (const float*)d_in[22];
  const float* gn2_w   = (const float*)d_in[23];
  const float* gn2_b   = (const float*)d_in[24];
  const float* gn2_ms  = (const float*)d_in[25];
  const float* ln1_g   = (const float*)d_in[26];
  const float* ln1_b   = (const float*)d_in[27];
  const float* ln2_g   = (const float*)d_in[28];
  const float* ln2_b   = (const float*)d_in[29];
  const float* ffn1_w  = (const float*)d_in[30];
  const float* ffn1_b  = (const float*)d_in[31];
  const float* ffn2_w  = (const float*)d_in[32];
  const float* ffn2_b  = (const float*)d_in[33];

  const int N = in_sizes[0] / DFEAT;
  const int E = in_sizes[1];
  const size_t ND = (size_t)N * DFEAT;

  // workspace layout (5 reused N*D slots + small buffers) ~262 MB
  float* ws   = (float*)d_ws;
  float* hn   = ws;            // normalized input, live until attention done
  float* buf1 = ws + ND;       // Q  -> hO   -> ffn_t(lo half)
  float* buf2 = ws + 2 * ND;   // K  -> h1   -> ffn_t(hi half)
  float* buf3 = ws + 3 * ND;   // V  -> sub2 -> h3
  float* buf4 = ws + 4 * ND;   // sub1 / S_r / wV / attn -> h2
  float* Wq     = ws + 5 * ND;                   // 9*128*128
  float* Wk     = Wq + NREL * DFEAT * DFEAT;
  float* Wv     = Wk + NREL * DFEAT * DFEAT;
  float* sums   = Wv + NREL * DFEAT * DFEAT;     // G*D
  float* varsum = sums + NGRAPHS * DFEAT;        // G*D
  float* counts = varsum + NGRAPHS * DFEAT;      // G
  float* z      = counts + NGRAPHS;              // N*HEADS

  const dim3 blk(256);
  const int ndThreads = N * DFEAT;
  const dim3 ndGrid((ndThreads + 255) / 256);
  const dim3 gemmGrid((N + 15) / 16, 1);
  const dim3 gemmGrid2((N + 15) / 16, 2);
  const int wElems = NREL * DFEAT * DFEAT;
  const size_t statBytes = (size_t)(2 * NGRAPHS * DFEAT + NGRAPHS) * sizeof(float);

  // --- basis-composed relation matrices ---
  build_W<<<(wElems + 255) / 256, blk, 0, stream>>>(q_coeff, q_basis, Wq);
  build_W<<<(wElems + 255) / 256, blk, 0, stream>>>(k_coeff, k_basis, Wk);
  build_W<<<(wElems + 255) / 256, blk, 0, stream>>>(v_coeff, v_basis, Wv);

  // --- graphNorm 1: h -> hn ---
  hipMemsetAsync(sums, 0, statBytes, stream);
  gn_stats<<<ndGrid, blk, 0, stream>>>(h, seg, sums, counts, N);
  gn_sub<<<ndGrid, blk, 0, stream>>>(h, seg, sums, counts, gn1_ms, buf4, varsum, N);
  gn_fin<<<ndGrid, blk, 0, stream>>>(buf4, seg, varsum, counts, gn1_w, gn1_b, hn, N);

  // --- relconv Q/K/V: init with self-loop term + bias ---
  gemm_wmma<false, true, false><<<gemmGrid, blk, 0, stream>>>(hn, q_loop, q_bias, buf1, N, DFEAT, DFEAT);
  gemm_wmma<false, true, false><<<gemmGrid, blk, 0, stream>>>(hn, k_loop, k_bias, buf2, N, DFEAT, DFEAT);
  gemm_wmma<false, true, false><<<gemmGrid, blk, 0, stream>>>(hn, v_loop, v_bias, buf3, N, DFEAT, DFEAT);

  // --- per relation: scatter S_r, then 3x GEMM-accumulate (relu fused in last) ---
  const dim3 scGrid(((size_t)E * 32 + 255) / 256);
  for (int r = 0; r < NREL; ++r) {
    hipMemsetAsync(buf4, 0, ND * sizeof(float), stream);
    scatter_rel<<<scGrid, blk, 0, stream>>>(hn, src, dst, etypes, r, buf4, E);
    const float* wqr = Wq + (size_t)r * DFEAT * DFEAT;
    const float* wkr = Wk + (size_t)r * DFEAT * DFEAT;
    const float* wvr = Wv + (size_t)r * DFEAT * DFEAT;
    if (r < NREL - 1) {
      gemm_wmma<true, false, false><<<gemmGrid, blk, 0, stream>>>(buf4, wqr, nullptr, buf1, N, DFEAT, DFEAT);
      gemm_wmma<true, false, false><<<gemmGrid, blk, 0, stream>>>(buf4, wkr, nullptr, buf2, N, DFEAT, DFEAT);
      gemm_wmma<true, false, false><<<gemmGrid, blk, 0, stream>>>(buf4, wvr, nullptr, buf3, N, DFEAT, DFEAT);
    } else {  // last relation: fuse final relu
      gemm_wmma<true, false, true><<<gemmGrid, blk, 0, stream>>>(buf4, wqr, nullptr, buf1, N, DFEAT, DFEAT);
      gemm_wmma<true, false, true><<<gemmGrid, blk, 0, stream>>>(buf4, wkr, nullptr, buf2, N, DFEAT, DFEAT);
      gemm_wmma<true, false, true><<<gemmGrid, blk, 0, stream>>>(buf4, wvr, nullptr, buf3, N, DFEAT, DFEAT);
    }
  }

  // --- edge attention -> attn (in-place in buf4) ---
  hipMemsetAsync(buf4, 0, ND * sizeof(float), stream);
  hipMemsetAsync(z, 0, (size_t)N * NHEADS * sizeof(float), stream);
  edge_attn<<<((size_t)E * NHEADS + 255) / 256, blk, 0, stream>>>(buf1, buf2, buf3, src, dst, buf4, z, E);
  attn_norm<<<ndGrid, blk, 0, stream>>>(buf4, z, N);

  // --- hO = attn @ o_w + o_b -> buf1 ; h1 = LN(hO) -> buf2 ---
  gemm_wmma<false, true, false><<<gemmGrid, blk, 0, stream>>>(buf4, o_w, o_b, buf1, N, DFEAT, DFEAT);
  layer_norm_k<<<(N + 7) / 8, blk, 0, stream>>>(buf1, ln1_g, ln1_b, buf2, N);

  // --- graphNorm 2: h1 -> h2 (buf4) ---
  hipMemsetAsync(sums, 0, statBytes, stream);
  gn_stats<<<ndGrid, blk, 0, stream>>>(buf2, seg, sums, counts, N);
  gn_sub<<<ndGrid, blk, 0, stream>>>(buf2, seg, sums, counts, gn2_ms, buf3, varsum, N);
  gn_fin<<<ndGrid, blk, 0, stream>>>(buf3, seg, varsum, counts, gn2_w, gn2_b, buf4, N);

  // --- FFN: relu(h2 @ W1 + b1) [N,256] -> buf1..buf2 ; @ W2 + b2 -> buf3 ---
  gemm_wmma<false, true, true><<<gemmGrid2, blk, 0, stream>>>(buf4, ffn1_w, ffn1_b, buf1, N, DFEAT, 2 * DFEAT);
  gemm_wmma<false, true, false><<<gemmGrid, blk, 0, stream>>>(buf1, ffn2_w, ffn2_b, buf3, N, 2 * DFEAT, DFEAT);

  // --- final LayerNorm -> d_out ---
  layer_norm_k<<<(N + 7) / 8, blk, 0, stream>>>(buf3, ln2_g, ln2_b, (float*)d_out, N);
}